// StructuralFingerprintLayer_8074538517086
// MI455X (gfx1250) — compile-verified
//
#include <hip/hip_runtime.h>
#include <hip/hip_bf16.h>

typedef float v2f __attribute__((ext_vector_type(2)));
typedef float v8f __attribute__((ext_vector_type(8)));

#define N_NODES 8192
#define F_IN    1024
#define F_OUT   256
#define NEGINF  (-9e15f)

// ---------------------------------------------------------------------------
// Kernel 1: H = X @ W   [8192,1024] x [1024,256], f32 WMMA 16x16x4.
// One 16x16 output tile per wave. 8 waves / block, 1024 blocks = 8192 tiles.
// A frag (16x4 f32): lane<16 holds M=lane, K={0,1}; lane>=16 holds K={2,3}.
// B frag (4x16 f32): lane<16 holds K=0/1 rows, lane>=16 K=2/3 rows, N=lane&15.
// ---------------------------------------------------------------------------
__global__ __launch_bounds__(256) void gemm_xw_kernel(const float* __restrict__ X,
                                                      const float* __restrict__ W,
                                                      float* __restrict__ H) {
  const int lane = threadIdx.x & 31;
  const int wave = threadIdx.x >> 5;
  const int gt   = blockIdx.x * 8 + wave;   // global tile id, 8192 total
  const int tm   = gt >> 4;                 // 0..511 (M tiles)
  const int tn   = gt & 15;                 // 0..15  (N tiles)
  const int half = lane >> 4;
  const int l16  = lane & 15;

  const float* aptr = X + (tm * 16 + l16) * F_IN + half * 2;
  const float* bptr = W + half * 2 * F_OUT + (tn * 16 + l16);

  v8f acc = {};
  for (int k = 0; k < F_IN; k += 4) {
    v2f a = *(const v2f*)(aptr + k);
    v2f b;
    b.x = bptr[k * F_OUT];
    b.y = bptr[k * F_OUT + F_OUT];
    acc = __builtin_amdgcn_wmma_f32_16x16x4_f32(false, a, false, b,
                                                (short)0, acc, false, false);
  }
  float* out = H + (tm * 16 + half * 8) * F_OUT + tn * 16 + l16;
#pragma unroll
  for (int m = 0; m < 8; ++m) out[m * F_OUT] = acc[m];
}

// ---------------------------------------------------------------------------
// Kernel 2: e_src[i] = h[i,:]·a[:256], e_dst[i] = h[i,:]·a[256:]. Wave per row.
// ---------------------------------------------------------------------------
__global__ __launch_bounds__(256) void attvec_kernel(const float* __restrict__ H,
                                                     const float* __restrict__ a,
                                                     float* __restrict__ esrc,
                                                     float* __restrict__ edst) {
  const int row  = (blockIdx.x * blockDim.x + threadIdx.x) >> 5;
  const int lane = threadIdx.x & 31;
  if (row >= N_NODES) return;
  const float* hrow = H + row * F_OUT;
  float s = 0.f, d = 0.f;
#pragma unroll
  for (int f = lane; f < F_OUT; f += 32) {
    float hv = hrow[f];
    s += hv * a[f];
    d += hv * a[F_OUT + f];
  }
#pragma unroll
  for (int m = 16; m; m >>= 1) {
    s += __shfl_xor(s, m, 32);
    d += __shfl_xor(d, m, 32);
  }
  if (lane == 0) { esrc[row] = s; edst[row] = d; }
}

// ---------------------------------------------------------------------------
// Kernel 3: fused masked softmax attention + (P @ H) + ELU, online softmax.
// Block = 8 waves = 256 threads, 64 i-rows per workgroup (128 blocks).
// Per 128-j step: each wave builds one 16-col masked-logit tile for all 64
// rows (4 row-tiles, WMMA C layout in registers), row-max via 16-lane
// shfl_xor reductions, p-tiles staged in LDS (row pad 18 -> aligned b64
// A-frag reads), accumulators rescaled, then each wave runs
// 8(q) x 4(kk) x 4(rt) x 2(t) = 256 f32 WMMAs; every H B-fragment pulled
// from L2 is reused by 4 row-tiles, cutting L2 traffic for H to ~1 GB.
// ---------------------------------------------------------------------------
__global__ __launch_bounds__(256) void gat_attn_kernel(
    const int* __restrict__ adj, const float* __restrict__ adj_ad,
    const float* __restrict__ H, const float* __restrict__ esrc,
    const float* __restrict__ edst, const float* __restrict__ Wei,
    const float* __restrict__ Wsi, float* __restrict__ out) {
  const int lane = threadIdx.x & 31;
  const int w    = threadIdx.x >> 5;   // wave 0..7
  const int half = lane >> 4;
  const int l16  = lane & 15;
  const int i0   = blockIdx.x * 64;

  __shared__ float pt[8][64][18];   // p tiles, padded for aligned b64 reads
  __shared__ float tmax[8][64];     // per-wave-tile row maxima
  __shared__ float psums[8][64];    // per-wave-tile row sums of p
  __shared__ float rowmax[64];
  __shared__ float rowsum[64];
  __shared__ float esb[64];

  if (threadIdx.x < 64) {
    rowmax[threadIdx.x] = -__builtin_inff();
    rowsum[threadIdx.x] = 0.f;
    esb[threadIdx.x]    = esrc[i0 + threadIdx.x];
  }

  const float wei = fabsf(Wei[0]);
  const float wsi = fabsf(Wsi[0]);

  v8f acc[4][2];
#pragma unroll
  for (int rt = 0; rt < 4; ++rt) {
    acc[rt][0] = (v8f){};
    acc[rt][1] = (v8f){};
  }

  const int colB = w * 32 + l16;   // this wave's output column base (+t*16)

  for (int j0 = 0; j0 < N_NODES; j0 += 128) {
    __syncthreads();  // B0: previous WMMA reads of pt done; init/esb visible
    const int jc = j0 + w * 16 + l16;   // this lane's j column
    const float ed = edst[jc];

    // ---- Phase A: masked logit tiles (C layout), per-row tile maxima -----
    float et[4][8];
#pragma unroll
    for (int rt = 0; rt < 4; ++rt) {
      const long base = (long)(i0 + rt * 16 + half * 8) * N_NODES + jc;
      if (j0 + 128 < N_NODES) {   // prefetch next tile of the HBM stream
        __builtin_prefetch((const void*)(adj + base + 128), 0, 0);
        __builtin_prefetch((const void*)(adj_ad + base + 128), 0, 0);
      }
#pragma unroll
      for (int m = 0; m < 8; ++m) {
        const long off = base + (long)m * N_NODES;
        float e = esb[rt * 16 + half * 8 + m] + ed;
        e = e > 0.f ? e : 0.2f * e;                  // leaky relu
        e = wei * e + wsi * adj_ad[off];
        et[rt][m] = adj[off] > 0 ? e : NEGINF;
      }
#pragma unroll
      for (int m = 0; m < 8; ++m) {
        float v = et[rt][m];
        v = fmaxf(v, __shfl_xor(v, 1, 32));
        v = fmaxf(v, __shfl_xor(v, 2, 32));
        v = fmaxf(v, __shfl_xor(v, 4, 32));
        v = fmaxf(v, __shfl_xor(v, 8, 32));
        if (l16 == 0) tmax[w][rt * 16 + half * 8 + m] = v;
      }
    }
    __syncthreads();  // B1

    // ---- Phase B: new running max, rescale acc, p tiles -> LDS -----------
#pragma unroll
    for (int rt = 0; rt < 4; ++rt) {
#pragma unroll
      for (int m = 0; m < 8; ++m) {
        const int r = rt * 16 + half * 8 + m;
        const float om = rowmax[r];
        float x = om;
#pragma unroll
        for (int qq = 0; qq < 8; ++qq) x = fmaxf(x, tmax[qq][r]);
        const float sc = __expf(om - x);
        acc[rt][0][m] *= sc;
        acc[rt][1][m] *= sc;
        const float p = __expf(et[rt][m] - x);
        pt[w][r][l16] = p;
        float v = p;
        v += __shfl_xor(v, 1, 32);
        v += __shfl_xor(v, 2, 32);
        v += __shfl_xor(v, 4, 32);
        v += __shfl_xor(v, 8, 32);
        if (l16 == 0) psums[w][r] = v;
      }
    }
    __syncthreads();  // B2

    // ---- Phase C: waves 0-1 update shared running stats ------------------
    if (threadIdx.x < 64) {
      const int r = threadIdx.x;
      const float om = rowmax[r];
      float x = om;
#pragma unroll
      for (int qq = 0; qq < 8; ++qq) x = fmaxf(x, tmax[qq][r]);
      float s = rowsum[r] * __expf(om - x);
#pragma unroll
      for (int qq = 0; qq < 8; ++qq) s += psums[qq][r];
      rowmax[r] = x;
      rowsum[r] = s;
    }

    // ---- Phase D: acc += P(64x128) @ H(128 x 32-slice), 256 WMMAs --------
#pragma unroll
    for (int q = 0; q < 8; ++q) {
#pragma unroll
      for (int kk = 0; kk < 4; ++kk) {
        const float* hb = H + (long)(j0 + q * 16 + kk * 4 + half * 2) * F_OUT;
        v2f b0, b1;
        b0.x = hb[colB];        b0.y = hb[F_OUT + colB];
        b1.x = hb[colB + 16];   b1.y = hb[F_OUT + colB + 16];
#pragma unroll
        for (int rt = 0; rt < 4; ++rt) {
          const v2f af = *(const v2f*)&pt[q][rt * 16 + l16][kk * 4 + half * 2];
          acc[rt][0] = __builtin_amdgcn_wmma_f32_16x16x4_f32(false, af, false, b0,
                                                             (short)0, acc[rt][0], false, false);
          acc[rt][1] = __builtin_amdgcn_wmma_f32_16x16x4_f32(false, af, false, b1,
                                                             (short)0, acc[rt][1], false, false);
        }
      }
    }
  }
  __syncthreads();  // final rowsum visible

  // ---- Epilogue: normalize + ELU, store ----------------------------------
#pragma unroll
  for (int rt = 0; rt < 4; ++rt) {
#pragma unroll
    for (int m = 0; m < 8; ++m) {
      const int r = rt * 16 + half * 8 + m;
      const float inv = 1.0f / rowsum[r];
#pragma unroll
      for (int t = 0; t < 2; ++t) {
        float v = acc[rt][t][m] * inv;
        v = v > 0.f ? v : (__expf(v) - 1.0f);     // ELU
        out[(long)(i0 + r) * F_OUT + colB + t * 16] = v;
      }
    }
  }
}

// ---------------------------------------------------------------------------
extern "C" void kernel_launch(void* const* d_in, const int* in_sizes, int n_in,
                              void* d_out, int out_size, void* d_ws, size_t ws_size,
                              hipStream_t stream) {
  const float* inp    = (const float*)d_in[0];
  const int*   adj    = (const int*)  d_in[1];
  const float* adj_ad = (const float*)d_in[2];
  const float* W      = (const float*)d_in[3];
  const float* a      = (const float*)d_in[4];
  const float* Wei    = (const float*)d_in[5];
  const float* Wsi    = (const float*)d_in[6];
  float* out = (float*)d_out;

  float* H    = (float*)d_ws;                 // 8192*256 f32 = 8 MB
  float* esrc = H + (size_t)N_NODES * F_OUT;  // 8192 f32
  float* edst = esrc + N_NODES;               // 8192 f32

  gemm_xw_kernel <<<1024, 256, 0, stream>>>(inp, W, H);
  attvec_kernel  <<<1024, 256, 0, stream>>>(H, a, esrc, edst);
  gat_attn_kernel<<<128,  256, 0, stream>>>(adj, adj_ad, H, esrc, edst, Wei, Wsi, out);
}